// IsingModel_16965120819509
// MI455X (gfx1250) — compile-verified
//
#include <hip/hip_runtime.h>
#include <cstdint>

// Problem constants (reference: R=10, S=20, N=360, BETA=0.5)
constexpr int   kN           = 360;
constexpr int   kTiles       = 23;                // ceil(360/16)
constexpr float kInvBetaNeg2 = -1.0f;             // 1/(-2*0.5)

typedef __attribute__((ext_vector_type(2))) float v2f;
typedef __attribute__((ext_vector_type(8))) float v8f;

// Broadcast a float from a (uniform) source lane via v_readlane_b32.
__device__ __forceinline__ float lane_bcast(float v, int srclane) {
  return __int_as_float(__builtin_amdgcn_readlane(__float_as_int(v), srclane));
}

__global__ __launch_bounds__(256) void ising_sweep_kernel(
    const float* __restrict__ s_in,
    const float* __restrict__ h,
    const float* __restrict__ J,
    const float* __restrict__ u,
    float* __restrict__ s_out)
{
  __shared__ float lds_s[kN];
  __shared__ float lds_field[kTiles * 16];   // 368 (8 tail slots are scratch)

  const int    rs    = blockIdx.x;           // which (r,s) chain
  const int    tid   = threadIdx.x;
  const int    lane  = tid & 31;
  const int    wid   = tid >> 5;
  const size_t vbase = (size_t)rs * kN;
  const float* Jrs   = J + (size_t)rs * kN * kN;

  // Stage spins into LDS (broadcast source for WMMA A operand).
  for (int t = tid; t < kN; t += blockDim.x) lds_s[t] = s_in[vbase + t];
  __syncthreads();

  // ---------------- Phase 1: field0 = J^T * s via V_WMMA_F32_16X16X4_F32 ----
  // D(16x16) = A(16x4) x B(4x16) + C, accumulated over K-chunks of 4.
  //   A[i][k3] = s[m0+k3]            (replicated over all 16 rows i)
  //   B[k3][n] = J[m0+k3][kt*16+n]   (contiguous row segments)
  // => D[i][n] = partial field[kt*16+n], identical for every row i.
  // Tail tile: column index is CLAMPED (not predicated) — B columns are
  // independent in the matmul, so out-of-range lanes just compute duplicate
  // values that land in unread LDS slots. No exec-mask branching needed.
  {
    const int col   = lane & 15;            // n within tile
    const int half  = lane >> 4;            // 0: K rows {0,1}; 1: K rows {2,3}
    for (int kt = wid; kt < kTiles; kt += 8) {
      int kcol = kt * 16 + col;
      if (kcol > kN - 1) kcol = kN - 1;     // clamp: stay in-bounds, branch-free
      const float* Bcol = Jrs + kcol;
      v8f c0 = {0.f,0.f,0.f,0.f,0.f,0.f,0.f,0.f};
      v8f c1 = {0.f,0.f,0.f,0.f,0.f,0.f,0.f,0.f};
      for (int m0 = 0; m0 < kN; m0 += 8) {  // 45 iters, 2 independent WMMAs
        const int kr0 = m0 + half * 2;
        const int kr1 = m0 + 4 + half * 2;
        v2f a0, a1, b0, b1;
        a0.x = lds_s[kr0];     a0.y = lds_s[kr0 + 1];
        a1.x = lds_s[kr1];     a1.y = lds_s[kr1 + 1];
        b0.x = Bcol[(size_t)kr0 * kN];
        b0.y = Bcol[(size_t)(kr0 + 1) * kN];
        b1.x = Bcol[(size_t)kr1 * kN];
        b1.y = Bcol[(size_t)(kr1 + 1) * kN];
        c0 = __builtin_amdgcn_wmma_f32_16x16x4_f32(
                false, a0, false, b0, (short)0, c0, false, false);
        c1 = __builtin_amdgcn_wmma_f32_16x16x4_f32(
                false, a1, false, b1, (short)0, c1, false, false);
      }
      // C/D layout: VGPR0 on lanes 0-15 is (M=0, N=lane) -> field[kt*16+lane]
      if (lane < 16) lds_field[kt * 16 + lane] = c0[0] + c1[0];
    }
  }
  __syncthreads();

  if (wid != 0) return;   // sweep is single-wave: lockstep, barrier-free

  // ---------------- Phase 2: sequential sweep, incremental field update ----
  // Lane owns k = lane + 32*i. All state in registers.
  float f[12], sl[12], rb[12];
  #pragma unroll
  for (int i = 0; i < 12; ++i) {
    const int k = lane + 32 * i;
    if (k < kN) {
      f[i]  = lds_field[k] + h[vbase + k];
      sl[i] = lds_s[k];
      rb[i] = kInvBetaNeg2 * logf(u[vbase + k]);   // rand_choice
    } else {
      f[i] = 0.f; sl[i] = 0.f; rb[i] = 0.f;
    }
  }

  #pragma unroll
  for (int i = 0; i < 12; ++i) {                   // register index: constant
    const int jcount = (i == 11) ? 8 : 32;
    for (int jj = 0; jj < jcount; ++jj) {          // serial spins within group
      const int   j  = i * 32 + jj;
      const float fj = lane_bcast(f[i],  jj);      // field_j  (owner lane jj)
      const float sj = lane_bcast(sl[i], jj);      // s_j
      const float rj = lane_bcast(rb[i], jj);      // r_j
      // Prefetch next J row while the flip decision resolves.
      if (j + 1 < kN)
        __builtin_prefetch(Jrs + (size_t)(j + 1) * kN + lane * 12, 0, 1);
      const bool flip = rj > sj * fj;              // uniform across the wave
      if (flip) {                                  // skip row read if no flip
        if (lane == jj) sl[i] = -sj;
        const float  delta = -2.0f * sj;           // s_new - s_old
        const float* row   = Jrs + (size_t)j * kN; // J[j][:], contiguous
        #pragma unroll
        for (int t = 0; t < 12; ++t) {
          if (t < 11 || lane < 8)                  // k = lane + 32*t < 360
            f[t] += row[lane + 32 * t] * delta;    // field_k += J[j][k]*delta
        }
      }
    }
  }

  #pragma unroll
  for (int i = 0; i < 12; ++i) {
    const int k = lane + 32 * i;
    if (k < kN) s_out[vbase + k] = sl[i];
  }
}

extern "C" void kernel_launch(void* const* d_in, const int* in_sizes, int n_in,
                              void* d_out, int out_size, void* d_ws, size_t ws_size,
                              hipStream_t stream) {
  (void)n_in; (void)out_size; (void)d_ws; (void)ws_size;
  const float* s = (const float*)d_in[0];   // (R,S,N) float32
  const float* h = (const float*)d_in[1];   // (R,S,N) float32
  const float* J = (const float*)d_in[2];   // (R,S,N,N) float32
  const float* u = (const float*)d_in[3];   // (R,S,N) float32
  float* out = (float*)d_out;               // (R,S,N) float32
  const int RS = in_sizes[0] / kN;          // 200 chains
  ising_sweep_kernel<<<dim3(RS), dim3(256), 0, stream>>>(s, h, J, u, out);
}